// MemoryBank_9552007266592
// MI455X (gfx1250) — compile-verified
//
#include <hip/hip_runtime.h>
#include <hip/hip_bf16.h>
#include <math.h>

// Problem constants (D and K are structural; B/N/P/G derived from in_sizes)
#define DIM    128
#define KTOP   32
#define QSTR   132   // padded LDS row stride (dwords): 132 % 64 = 4 -> conflict-free 16-row access
#define NEGV   -1000000000.0f
#define GSPLIT 4     // blockIdx.z partitions of the candidate range (occupancy)

typedef __attribute__((ext_vector_type(2))) float v2f;
typedef __attribute__((ext_vector_type(8))) float v8f;

// ---------------------------------------------------------------- phase 1a
__global__ __launch_bounds__(64) void zero_counts_kernel(int* __restrict__ qcount, int P) {
    int t = threadIdx.x + blockIdx.x * blockDim.x;
    if (t < P) qcount[t] = 0;
}

// ---------------------------------------------------------------- phase 1b
__global__ __launch_bounds__(256) void build_lists_kernel(const int* __restrict__ hard,
                                                          int* __restrict__ qcount,
                                                          int* __restrict__ qlist, int B) {
    int b = threadIdx.x + blockIdx.x * blockDim.x;
    if (b < B) {
        int p = hard[b];
        int slot = atomicAdd(&qcount[p], 1);
        qlist[(size_t)p * B + slot] = b;
    }
}

// ---------------------------------------------------------------- phase 2
// grid = (P, ceil(B/16), GSPLIT); block = 128 (4 waves).
// Block z owns candidates [z*cpb, (z+1)*cpb); wave w owns cpw = cpb/4 of them,
// processed as chunks of 16 rows -> one 16x16 WMMA tile per chunk (K=128).
__global__ __launch_bounds__(128) void sim_wmma_kernel(
    const float* __restrict__ query, const float* __restrict__ keys,
    const long long* __restrict__ cg, const int* __restrict__ qlist,
    const int* __restrict__ qcount, float* __restrict__ sim,
    int B, int N, int G, int cpb)
{
    const int p    = blockIdx.x;
    const int tile = blockIdx.y;
    const int z    = blockIdx.z;
    const int cnt  = qcount[p];
    if (tile * 16 >= cnt) return;                 // uniform per block
    const int m_count = min(16, cnt - tile * 16);

    const int tid  = threadIdx.x;
    const int wave = tid >> 5;
    const int lane = tid & 31;
    const int chunks_pw = cpb >> 6;               // (cpb/4 waves)/16 rows

    __shared__ __align__(16) float Qs[16 * QSTR];
    __shared__ __align__(16) float Ks[4][16 * QSTR];
    __shared__ float invq[16];
    __shared__ float invk[4][16];
    __shared__ int   qid[16];
    __shared__ int   cand_s[4][16];
    __shared__ int   valid_s[4][16];

    if (tid < 16)
        qid[tid] = (tid < m_count) ? qlist[(size_t)p * B + tile * 16 + tid] : 0;
    __syncthreads();

    // cooperative load of the 16 x 128 query tile (zeros for padding rows)
    for (int e = tid; e < 16 * DIM; e += 128) {
        int m = e >> 7, c = e & (DIM - 1);
        float v = 0.f;
        if (m < m_count) v = query[(size_t)qid[m] * DIM + c];
        Qs[m * QSTR + c] = v;
    }
    __syncthreads();

    if (tid < 16) {
        float s = 0.f;
        for (int c = 0; c < DIM; ++c) { float x = Qs[tid * QSTR + c]; s += x * x; }
        invq[tid] = 1.f / fmaxf(sqrtf(s), 1e-12f);   // F.normalize eps semantics
    }
    __syncthreads();

    const int n16 = lane & 15;
    const int hi  = lane >> 4;    // selects K-pair (0,1) vs (2,3) per WMMA layout

    for (int cc = 0; cc < chunks_pw; ++cc) {
        const int g0 = z * cpb + (wave * chunks_pw + cc) * 16;

        if (lane < 16) {
            long long c = cg[(size_t)p * G + g0 + lane];
            valid_s[wave][lane] = (c >= 0) ? 1 : 0;
            long long sc = (c < 0) ? 0 : c;
            if (sc > (long long)N - 1) sc = N - 1;
            cand_s[wave][lane] = (int)sc;
        }
        __syncthreads();

        // gather 16 key rows into LDS; wave-reduce sum-of-squares per row
        for (int r = 0; r < 16; ++r) {
            const float4* src = reinterpret_cast<const float4*>(
                keys + (size_t)cand_s[wave][r] * DIM);
            float4 kv = src[lane];                              // 32 lanes x 16B = 512B row
            *reinterpret_cast<float4*>(&Ks[wave][r * QSTR + lane * 4]) = kv;
            float s = kv.x * kv.x + kv.y * kv.y + kv.z * kv.z + kv.w * kv.w;
            for (int off = 16; off > 0; off >>= 1) s += __shfl_xor(s, off, 32);
            if (lane == 0) invk[wave][r] = 1.f / fmaxf(sqrtf(s), 1e-12f);
        }
        __syncthreads();

        // 16x16 sim tile: 32 chained V_WMMA_F32_16X16X4_F32 (K = 128)
        v8f acc = {0.f, 0.f, 0.f, 0.f, 0.f, 0.f, 0.f, 0.f};
        for (int s = 0; s < 32; ++s) {
            const int kb = s * 4 + hi * 2;
            v2f a, bf;
            a[0]  = Qs[n16 * QSTR + kb];
            a[1]  = Qs[n16 * QSTR + kb + 1];
            bf[0] = Ks[wave][n16 * QSTR + kb];
            bf[1] = Ks[wave][n16 * QSTR + kb + 1];
            acc = __builtin_amdgcn_wmma_f32_16x16x4_f32(
                false, a, false, bf, (short)0, acc, false, false);
        }

        // post-scale by both inverse norms; mask invalid; scatter to sim[B][G]
        const float ik    = invk[wave][n16];
        const int   valid = valid_s[wave][n16];
        const int   g     = g0 + n16;
        for (int v = 0; v < 8; ++v) {
            int m = v + hi * 8;                   // C/D layout: row = vgpr + 8*(lane>=16)
            if (m < m_count) {
                float val = valid ? acc[v] * invq[m] * ik : NEGV;
                sim[(size_t)qid[m] * G + g] = val;
            }
        }
        __syncthreads();                          // protect Ks before next chunk
    }
}

// ---------------------------------------------------------------- phase 3
// One block (256 threads = 8 waves) per query. Iterative top-32 with
// shuffle-based wave reduction (2 barriers per selection), jax.lax.top_k
// ordering (descending value, ties -> lower index), then gather + outputs.
__global__ __launch_bounds__(256) void topk_gather_kernel(
    const float* __restrict__ keys, const float* __restrict__ vals,
    const long long* __restrict__ cg, const int* __restrict__ hard,
    const float* __restrict__ sim, float* __restrict__ out,
    int B, int N, int G)
{
    const int b    = blockIdx.x;
    const int t    = threadIdx.x;
    const int wave = t >> 5;
    const int vpt  = G >> 8;                       // values per thread (G/256 <= 8)

    float lv[8];
    for (int q = 0; q < vpt; ++q) lv[q] = sim[(size_t)b * G + t + (q << 8)];

    __shared__ float wrv[8];
    __shared__ int   wrg[8];
    __shared__ int   winner;
    __shared__ int   topg[KTOP];

    for (int j = 0; j < KTOP; ++j) {
        // local best over this thread's values
        float bv = -INFINITY; int bg = 0x7fffffff;
        for (int q = 0; q < vpt; ++q) {
            int g = t + (q << 8);
            if (lv[q] > bv || (lv[q] == bv && g < bg)) { bv = lv[q]; bg = g; }
        }
        // intra-wave shuffle reduction (wave32 lockstep, no barrier)
        for (int off = 16; off > 0; off >>= 1) {
            float ov = __shfl_xor(bv, off, 32);
            int   og = __shfl_xor(bg, off, 32);
            if (ov > bv || (ov == bv && og < bg)) { bv = ov; bg = og; }
        }
        if ((t & 31) == 0) { wrv[wave] = bv; wrg[wave] = bg; }
        __syncthreads();
        if (t == 0) {
            float fv = wrv[0]; int fg = wrg[0];
            for (int w = 1; w < 8; ++w) {
                if (wrv[w] > fv || (wrv[w] == fv && wrg[w] < fg)) {
                    fv = wrv[w]; fg = wrg[w];
                }
            }
            winner  = fg;
            topg[j] = fg;
        }
        __syncthreads();
        const int g = winner;
        if (t == (g & 255)) lv[g >> 8] = -INFINITY;   // owner removes winner
    }

    __shared__ int ifin[KTOP];
    const int p = hard[b];
    if (t < KTOP) {
        int g = topg[t];
        long long c = cg[(size_t)p * G + g];
        long long sc = (c < 0) ? 0 : c;
        if (sc > (long long)N - 1) sc = N - 1;
        ifin[t] = (int)sc;
    }
    __syncthreads();

    const size_t KD = (size_t)KTOP * DIM;
    float* onk = out;
    float* onv = out + (size_t)B * KD;
    float* zs  = onv + (size_t)B * KD;
    float* ifo = zs  + (size_t)B * KTOP;

    for (int e = t; e < KTOP * DIM; e += 256) {
        int j = e >> 7, dd = e & (DIM - 1);
        int idx = ifin[j];
        onk[(size_t)b * KD + e] = keys[(size_t)idx * DIM + dd];
        onv[(size_t)b * KD + e] = vals[(size_t)idx * DIM + dd];
    }
    if (t < KTOP) {
        zs[(size_t)b * KTOP + t]  = 0.f;
        ifo[(size_t)b * KTOP + t] = (float)ifin[t];  // exact: N < 2^24
    }
}

// ---------------------------------------------------------------- launch
extern "C" void kernel_launch(void* const* d_in, const int* in_sizes, int n_in,
                              void* d_out, int out_size, void* d_ws, size_t ws_size,
                              hipStream_t stream) {
    const float*     query = (const float*)d_in[0];
    const float*     keys  = (const float*)d_in[1];
    const float*     vals  = (const float*)d_in[2];
    const long long* cg    = (const long long*)d_in[3];   // int64 per reference
    const int*       hard  = (const int*)d_in[5];

    const int B = in_sizes[0] / DIM;
    const int N = in_sizes[1] / DIM;
    const int P = in_sizes[4];
    const int G = in_sizes[3] / P;
    const int cpb = G / GSPLIT;                   // candidates per block (G=1024 -> 256)

    // workspace layout: sim[B*G] f32 | qlist[P*B] i32 | qcount[P] i32
    float* sim    = (float*)d_ws;
    int*   qlist  = (int*)((char*)d_ws + (size_t)B * G * sizeof(float));
    int*   qcount = qlist + (size_t)P * B;

    zero_counts_kernel<<<(P + 63) / 64, 64, 0, stream>>>(qcount, P);
    build_lists_kernel<<<(B + 255) / 256, 256, 0, stream>>>(hard, qcount, qlist, B);

    dim3 g2(P, (B + 15) / 16, GSPLIT);
    sim_wmma_kernel<<<g2, 128, 0, stream>>>(query, keys, cg, qlist, qcount, sim,
                                            B, N, G, cpb);

    topk_gather_kernel<<<B, 256, 0, stream>>>(keys, vals, cg, hard, sim,
                                              (float*)d_out, B, N, G);
}